// TestModel_31791347925401
// MI455X (gfx1250) — compile-verified
//
#include <hip/hip_runtime.h>
#include <stdint.h>

// Problem constants (from reference)
#define XS   2048   // x spatial size
#define HIN  1026   // resized input H/W  (SIZEWH + KWH - 1)
#define WIN_ 1026
#define IPITCH 1028 // padded row pitch of resized input (16B-aligned rows)
#define IROWS  1028 // padded rows per channel plane (rows 1026,1027 zeroed)
#define HO   1024
#define WO   1024
#define CCH  3
#define KK   9

// Deform tile config: 128 wide x 8 tall, 256 threads, 4 consecutive px/thread
#define TW    128
#define TH    8
#define WROWS 12    // TH + 4
#define WPAD  132   // TW + 4  (multiple of 4 -> b128 staging)
#define WVEC  (CCH * WROWS * WPAD / 4)   // 1188 vec4 slots

// ---------------------------------------------------------------------------
// Pass 1: horizontal antialiased triangle resize x(3,2048,2048)->tmp(3,2048,1026)
// JAX: u=(i+0.5)*(in/out)-0.5, kscale=in/out, w=max(0,1-|u-j|/kscale),
// normalized over in-range j only.
// ---------------------------------------------------------------------------
__global__ __launch_bounds__(256) void resizeH_kernel(const float* __restrict__ x,
                                                      float* __restrict__ tmp) {
    const int total = CCH * XS * WIN_;
    int i = blockIdx.x * blockDim.x + threadIdx.x;
    if (i >= total) return;
    int ocol = i % WIN_;
    int r    = (i / WIN_) % XS;
    int c    = i / (WIN_ * XS);

    const float ks     = (float)XS / (float)WIN_;   // 1.99610...
    const float inv_ks = (float)WIN_ / (float)XS;
    float u = ((float)ocol + 0.5f) * ks - 0.5f;
    int jlo = (int)ceilf(u - ks);
    int jhi = (int)floorf(u + ks);
    if (jlo < 0) jlo = 0;
    if (jhi > XS - 1) jhi = XS - 1;

    const float* row = x + ((size_t)c * XS + r) * XS;
    float acc = 0.0f, twt = 0.0f;
    for (int j = jlo; j <= jhi; ++j) {
        float w = fmaxf(0.0f, 1.0f - fabsf(u - (float)j) * inv_ks);
        twt += w;
        acc += w * row[j];
    }
    tmp[i] = acc / twt;
}

// ---------------------------------------------------------------------------
// Pass 2: vertical resize tmp(3,2048,1026) -> inp(3,1028,1028) padded layout.
// Pad rows/cols (>=1026) are written as 0 so the deform kernel can stage
// unconditionally and OOB bilinear corners read exact zeros.
// ---------------------------------------------------------------------------
__global__ __launch_bounds__(256) void resizeV_kernel(const float* __restrict__ tmp,
                                                      float* __restrict__ inp) {
    const int total = CCH * IROWS * IPITCH;
    int i = blockIdx.x * blockDim.x + threadIdx.x;
    if (i >= total) return;
    int col  = i % IPITCH;
    int orow = (i / IPITCH) % IROWS;
    int c    = i / (IPITCH * IROWS);

    if (col >= WIN_ || orow >= HIN) { inp[i] = 0.0f; return; }

    const float ks     = (float)XS / (float)HIN;
    const float inv_ks = (float)HIN / (float)XS;
    float u = ((float)orow + 0.5f) * ks - 0.5f;
    int jlo = (int)ceilf(u - ks);
    int jhi = (int)floorf(u + ks);
    if (jlo < 0) jlo = 0;
    if (jhi > XS - 1) jhi = XS - 1;

    const float* colp = tmp + (size_t)c * XS * WIN_ + col;
    float acc = 0.0f, twt = 0.0f;
    for (int j = jlo; j <= jhi; ++j) {
        float w = fmaxf(0.0f, 1.0f - fabsf(u - (float)j) * inv_ks);
        twt += w;
        acc += w * colp[(size_t)j * WIN_];
    }
    inp[i] = acc / twt;
}

// ---------------------------------------------------------------------------
// Main deformable conv kernel. Block = 128x8 output tile, 256 threads,
// 4 consecutive x-pixels per thread (float4 offset/mask loads + float4 store).
// 3x12x132 input window staged to LDS with async b128 DMA (ASYNCcnt path).
// ---------------------------------------------------------------------------
__global__ __launch_bounds__(256) void deform_kernel(const float* __restrict__ inp,
                                                     const float* __restrict__ weight,
                                                     const float* __restrict__ offset,
                                                     const float* __restrict__ mask,
                                                     float* __restrict__ out) {
    __shared__ __align__(16) float win[CCH * WROWS * WPAD];  // 4752 floats = 19 KB
    __shared__ float wgt[CCH * KK];

    const int tid = threadIdx.x;
    const int tx  = blockIdx.x * TW;
    const int ty  = blockIdx.y * TH;

    if (tid < CCH * KK) wgt[tid] = weight[tid];

    // ---- stage input window into LDS: unconditional async b128 copies
    const uint32_t wbase = (uint32_t)(size_t)(&win[0]);   // LDS byte offset
    for (int slot = tid; slot < WVEC; slot += 256) {
        int c   = slot / (WROWS * WPAD / 4);              // /396
        int rem = slot - c * (WROWS * WPAD / 4);
        int wy  = rem / (WPAD / 4);                       // /33
        int wq  = rem - wy * (WPAD / 4);
        const float* gp = inp + (size_t)c * (IROWS * IPITCH)
                              + (size_t)(ty + wy) * IPITCH + tx + 4 * wq;
        uint32_t lo = wbase + (uint32_t)slot * 16u;
        unsigned long long ga = (unsigned long long)(size_t)gp;
        asm volatile("global_load_async_to_lds_b128 %0, %1, off"
                     :: "v"(lo), "v"(ga) : "memory");
    }
    asm volatile("s_wait_asynccnt 0x0" ::: "memory");
    __syncthreads();

    // ---- compute: thread -> row r (0..7), 4 consecutive x at 4*(tid&31)
    const int lane = tid & 31;
    const int r    = tid >> 5;
    const int y    = ty + r;
    const int x0p  = tx + 4 * lane;
    const size_t HW = (size_t)HO * WO;
    const size_t pb = (size_t)y * WO + x0p;

    float a0 = 0.0f, a1 = 0.0f, a2 = 0.0f, a3 = 0.0f;

#pragma unroll
    for (int k = 0; k < KK; ++k) {
        const int ky = k / 3;
        const int kx = k - 3 * ky;
        const float4 oy4 = *(const float4*)(offset + (size_t)(2 * k)     * HW + pb);
        const float4 ox4 = *(const float4*)(offset + (size_t)(2 * k + 1) * HW + pb);
        const float4 m4  = *(const float4*)(mask   + (size_t)k           * HW + pb);

        const float oys[4] = {oy4.x, oy4.y, oy4.z, oy4.w};
        const float oxs[4] = {ox4.x, ox4.y, ox4.z, ox4.w};
        const float ms [4] = {m4.x,  m4.y,  m4.z,  m4.w};
        float accs[4];
#pragma unroll
        for (int j = 0; j < 4; ++j) {
            const int xj = x0p + j;
            const float py  = (float)(y + ky)  + oys[j];
            const float px  = (float)(xj + kx) + oxs[j];
            const float y0f = floorf(py);
            const float x0f = floorf(px);
            const float fy  = py - y0f;
            const float fx  = px - x0f;
            int wy0 = (int)y0f - ty;
            int wx0 = (int)x0f - tx;
            // offsets in [0,1) keep samples inside the window; clamp for safety
            wy0 = min(max(wy0, 0), WROWS - 2);
            wx0 = min(max(wx0, 0), WPAD - 2);

            const float gy1 = 1.0f - fy;
            const float gx1 = 1.0f - fx;
            const float w00 = gy1 * gx1;
            const float w01 = gy1 * fx;
            const float w10 = fy  * gx1;
            const float w11 = fy  * fx;
            const int   b   = wy0 * WPAD + wx0;

            float acc = 0.0f;
#pragma unroll
            for (int c = 0; c < CCH; ++c) {
                const float* wp = &win[c * (WROWS * WPAD) + b];
                const float s = wp[0] * w00 + wp[1] * w01 +
                                wp[WPAD] * w10 + wp[WPAD + 1] * w11;
                acc += wgt[c * KK + k] * s;
            }
            accs[j] = acc * ms[j];
        }
        a0 += accs[0]; a1 += accs[1]; a2 += accs[2]; a3 += accs[3];
    }

    float4 o;
    o.x = a0; o.y = a1; o.z = a2; o.w = a3;
    *(float4*)(out + pb) = o;
}

// ---------------------------------------------------------------------------
extern "C" void kernel_launch(void* const* d_in, const int* in_sizes, int n_in,
                              void* d_out, int out_size, void* d_ws, size_t ws_size,
                              hipStream_t stream) {
    const float* x      = (const float*)d_in[0];   // (1,3,2048,2048)
    const float* weight = (const float*)d_in[1];   // (1,3,3,3)
    const float* offset = (const float*)d_in[2];   // (1,18,1024,1024)
    const float* mask   = (const float*)d_in[3];   // (1,9,1024,1024)
    float* out = (float*)d_out;                    // (1,1,1024,1024)

    float* tmp = (float*)d_ws;                          // 3*2048*1026 floats (25.2 MB)
    float* inp = tmp + (size_t)CCH * XS * WIN_;         // 3*1028*1028 floats (12.7 MB)

    {
        const int total = CCH * XS * WIN_;
        resizeH_kernel<<<(total + 255) / 256, 256, 0, stream>>>(x, tmp);
    }
    {
        const int total = CCH * IROWS * IPITCH;
        resizeV_kernel<<<(total + 255) / 256, 256, 0, stream>>>(tmp, inp);
    }
    {
        dim3 grid(WO / TW, HO / TH);   // 8 x 128 tiles
        deform_kernel<<<grid, 256, 0, stream>>>(inp, weight, offset, mask, out);
    }
}